// FeaturePrototype_59038620451264
// MI455X (gfx1250) — compile-verified
//
#include <hip/hip_runtime.h>

typedef __attribute__((ext_vector_type(2))) float v2f;
typedef __attribute__((ext_vector_type(4))) float v4f;
typedef __attribute__((ext_vector_type(8))) float v8f;

#define B_DIM 128          // batch
#define K_CLS 100          // classes
#define K_PAD 112          // classes padded to multiple of 16
#define N_DIM 32768        // C*H*W = 128*16*16
#define NT4   8192         // N_DIM / 4
#define M_TILES 7          // K_PAD / 16
#define N_TILES 2048       // N_DIM / 16

// ---------------------------------------------------------------------------
// Kernel 1: per-batch argmax of logits (softmax is monotonic -> argmax of
// logits, strict '>' keeps the first max like jnp.argmax), build the one-hot
// matrix A[K_PAD][B_DIM] (fp32 0/1) and per-class counts.
// ---------------------------------------------------------------------------
__global__ void argmax_onehot_kernel(const float* __restrict__ logits,
                                     float* __restrict__ A,
                                     float* __restrict__ cnt,
                                     int* __restrict__ cls) {
  const int t = threadIdx.x;  // 128 threads
  // zero the one-hot matrix cooperatively
  for (int i = t; i < K_PAD * B_DIM; i += B_DIM) A[i] = 0.0f;
  // argmax over 100 logits for batch row t
  const float* row = logits + t * K_CLS;
  float best = row[0];
  int bi = 0;
  for (int k = 1; k < K_CLS; ++k) {
    float v = row[k];
    if (v > best) { best = v; bi = k; }
  }
  cls[t] = bi;
  __syncthreads();
  A[bi * B_DIM + t] = 1.0f;  // unique (row,col) per thread, no conflict
  __syncthreads();
  if (t < K_PAD) {
    int c = 0;
    for (int b = 0; b < B_DIM; ++b) c += (cls[b] == t) ? 1 : 0;
    cnt[t] = (float)c;
  }
}

// ---------------------------------------------------------------------------
// Kernel 2: prototypes = (OneHot[K_PAD x 128] @ X[128 x 32768]) / count.
// One wave per 16x16 output tile; reduction over B=128 as 32 chained
// v_wmma_f32_16x16x4_f32 ops. Exact in fp32 (A entries are 0/1).
//
// ISA 7.12.2 layouts (wave32):
//   A 16x4 f32:  lanes 0-15: M=lane, v0=K0,v1=K1 ; lanes 16-31: v0=K2,v1=K3
//   B 4x16 f32:  v0: K=0 (lanes 0-15, N=lane) / K=2 (lanes 16-31)
//                v1: K=1 / K=3
//   C/D 16x16:   VGPR r: M=r (lanes 0-15), M=r+8 (lanes 16-31), N=lane&15
// ---------------------------------------------------------------------------
__global__ void proto_wmma_kernel(const float* __restrict__ X,
                                  const float* __restrict__ A,
                                  const float* __restrict__ cnt,
                                  float* __restrict__ P) {
  const int wave = threadIdx.x >> 5;
  const int lane = threadIdx.x & 31;
  const int tile = blockIdx.x * 8 + wave;   // 1792 blocks * 8 waves = 14336
  const int mt = tile % M_TILES;
  const int nt = tile / M_TILES;
  const int m0 = mt * 16;
  const int n0 = nt * 16;
  const int hs  = lane >> 4;   // half-wave select (0: lanes 0-15, 1: 16-31)
  const int l15 = lane & 15;

  // per-lane base pointers for the A and B fragments
  const float* Ap = A + (m0 + l15) * B_DIM + hs * 2;          // row m, cols b0+2*hs
  const float* Bp = X + (size_t)(hs * 2) * N_DIM + n0 + l15;  // rows b0+2*hs, col n

  v8f acc = {};
  for (int b0 = 0; b0 < B_DIM; b0 += 4) {
    v2f a, b;
    a.x = Ap[b0];
    a.y = Ap[b0 + 1];
    b.x = Bp[(size_t)b0 * N_DIM];
    b.y = Bp[(size_t)(b0 + 1) * N_DIM];
    // 8 args: (neg_a, A, neg_b, B, c_mod, C, reuse_a, reuse_b)
    acc = __builtin_amdgcn_wmma_f32_16x16x4_f32(false, a, false, b,
                                                (short)0, acc, false, false);
  }

#pragma unroll
  for (int r = 0; r < 8; ++r) {
    const int row = m0 + r + hs * 8;
    if (row < K_CLS) {
      const float c = cnt[row];
      // match reference: where(cnt>0, sums / max(cnt,1), 0) -> true division
      P[(size_t)row * N_DIM + n0 + l15] = (c > 0.0f) ? acc[r] / c : 0.0f;
    }
  }
}

// ---------------------------------------------------------------------------
// Kernel 3: inter[i][j][n] = P[j][n] - P[i][n]. 1.31 GB streaming write =
// the roofline cost (~57 us at 23.3 TB/s). P (13 MB) stays resident in the
// 192 MB L2; output uses non-temporal b128 stores.
// ---------------------------------------------------------------------------
__global__ void inter_diff_kernel(const float* __restrict__ P,
                                  float* __restrict__ out) {
  const size_t idx = (size_t)blockIdx.x * blockDim.x + threadIdx.x; // float4 idx
  const size_t n4 = idx & (NT4 - 1);
  const size_t ij = idx >> 13;                 // NT4 = 2^13
  const int j = (int)(ij % K_CLS);
  const int i = (int)(ij / K_CLS);
  const v4f* Pj = (const v4f*)(P + (size_t)j * N_DIM);
  const v4f* Pi = (const v4f*)(P + (size_t)i * N_DIM);
  const v4f d = Pj[n4] - Pi[n4];
  __builtin_nontemporal_store(d, ((v4f*)out) + idx);
}

// ---------------------------------------------------------------------------
extern "C" void kernel_launch(void* const* d_in, const int* in_sizes, int n_in,
                              void* d_out, int out_size, void* d_ws, size_t ws_size,
                              hipStream_t stream) {
  const float* x      = (const float*)d_in[0];  // (128,128,16,16) f32
  const float* logits = (const float*)d_in[1];  // (128,100) f32
  float* out = (float*)d_out;                   // prototypes ++ inter matrix

  // workspace layout (< 60 KB)
  float* A   = (float*)d_ws;                    // K_PAD*B_DIM one-hot
  float* cnt = A + K_PAD * B_DIM;               // K_PAD counts
  int*   cls = (int*)(cnt + K_PAD);             // B_DIM argmax results

  argmax_onehot_kernel<<<1, B_DIM, 0, stream>>>(logits, A, cnt, cls);

  // 7*2048 = 14336 tiles, 8 waves (256 thr) per block -> 1792 blocks
  proto_wmma_kernel<<<(M_TILES * N_TILES) / 8, 256, 0, stream>>>(x, A, cnt, out);

  float* inter = out + (size_t)K_CLS * N_DIM;   // offset 3,276,800 floats
  // 100*100*8192 float4 elements = 81,920,000 -> 320,000 blocks of 256
  inter_diff_kernel<<<(K_CLS * K_CLS * NT4) / 256, 256, 0, stream>>>(out, inter);
}